// TTTLayer_5626407158223
// MI455X (gfx1250) — compile-verified
//
#include <hip/hip_runtime.h>
#include <hip/hip_bf16.h>
#include <math.h>
#include <stdint.h>

// Problem constants (B=4, T=4096, C=1024, RANK=8)
#define CDIM 1024
#define BT_ROWS 16384   // B*T
#define RANK 8

// GEMM tiling: block 128x128, BK=32, 8 waves as 4(M)x2(N), wave tile 32x64
#define BM 128
#define BN 128
#define BK 32
#define LDK 36          // padded LDS row stride (words); 144B = multiple of 16
#define LDB 132         // padded row stride for [k][n] B tile; 528B = mult of 16
#define GEMM_BLOCKS ((BT_ROWS / BM) * (CDIM / BN))   // 128*8 = 1024

// Scan chunking
#define NCH 32
#define CLEN 128        // 4096 / NCH

typedef __attribute__((ext_vector_type(2))) float v2f;
typedef __attribute__((ext_vector_type(8))) float v8f;

__device__ __forceinline__ v8f wmma4(v2f a, v2f b, v8f c) {
  // V_WMMA_F32_16X16X4_F32: full-precision fp32 matrix op (CDNA5)
  return __builtin_amdgcn_wmma_f32_16x16x4_f32(
      /*neg_a=*/false, a, /*neg_b=*/false, b,
      /*c_mod=*/(short)0, c, /*reuse_a=*/false, /*reuse_b=*/false);
}

// CDNA5 async global->LDS DMA (ASYNCcnt-tracked); 16B per lane.
// LDS address = low 32 bits of the flat shared pointer.
__device__ __forceinline__ void async_ld_b128(const float* g, const float* l) {
  unsigned loff = (unsigned)(uintptr_t)(const void*)l;
  asm volatile("global_load_async_to_lds_b128 %0, %1, off"
               :: "v"(loff), "v"(g)
               : "memory");
}
__device__ __forceinline__ void wait_async0() {
  asm volatile("s_wait_asynccnt 0x0" ::: "memory");
}

// ---------------------------------------------------------------------------
// LayerNorm: one block per row of 1024
// ---------------------------------------------------------------------------
__global__ __launch_bounds__(256) void ln_kernel(const float* __restrict__ x,
                                                 const float* __restrict__ g,
                                                 const float* __restrict__ b,
                                                 float* __restrict__ h) {
  __shared__ float red[256];
  const int row = blockIdx.x;
  const int tid = threadIdx.x;
  const float* xr = x + (size_t)row * CDIM;
  float* hr = h + (size_t)row * CDIM;
  float v[4];
  float s = 0.f;
#pragma unroll
  for (int i = 0; i < 4; ++i) { v[i] = xr[tid + 256 * i]; s += v[i]; }
  red[tid] = s; __syncthreads();
  for (int o = 128; o > 0; o >>= 1) { if (tid < o) red[tid] += red[tid + o]; __syncthreads(); }
  const float mean = red[0] * (1.0f / CDIM);
  __syncthreads();
  float ss = 0.f;
#pragma unroll
  for (int i = 0; i < 4; ++i) { float d = v[i] - mean; ss += d * d; }
  red[tid] = ss; __syncthreads();
  for (int o = 128; o > 0; o >>= 1) { if (tid < o) red[tid] += red[tid + o]; __syncthreads(); }
  const float rstd = rsqrtf(red[0] * (1.0f / CDIM) + 1e-5f);
#pragma unroll
  for (int i = 0; i < 4; ++i) {
    int c = tid + 256 * i;
    hr[c] = (v[i] - mean) * rstd * g[c] + b[c];
  }
}

// ---------------------------------------------------------------------------
// Y = X * W^T  (einsum 'btc,dc->btd').  MODE 0: plain store.
// MODE 1: Y = resid + (X*W^T) * scale[col]   (final Wo + LayerScale + residual)
// Double-buffered LDS, async global->LDS tile fetch, 1 barrier per K-tile.
// ---------------------------------------------------------------------------
template <int MODE>
__global__ __launch_bounds__(256) void gemm_nt_kernel(const float* __restrict__ X,
                                                      const float* __restrict__ W,
                                                      float* __restrict__ Y,
                                                      const float* __restrict__ resid,
                                                      const float* __restrict__ scale) {
  __shared__ __align__(16) float As[2][BM][LDK];
  __shared__ __align__(16) float Ws[2][BN][LDK];
  const int tid  = threadIdx.x;
  const int lane = tid & 31;
  const int wv   = tid >> 5;
  const int wm   = wv & 3;          // 4 M strips of 32 rows
  const int wn   = wv >> 2;         // 2 N strips of 64 cols
  const int bm   = blockIdx.x >> 3; // 8 n-blocks per m-block (1024/128)
  const int bn   = blockIdx.x & 7;
  const int m0 = bm * BM, n0 = bn * BN;
  const int mrowA = wm * 32 + (lane & 15);
  const int nbase = wn * 64 + (lane & 15);
  const int koff = (lane < 16) ? 0 : 2;  // A frag k-offset per ISA layout
  const int bka  = (lane < 16) ? 0 : 1;  // B frag k-offset per ISA layout
  v8f acc[2][4] = {};

  auto issue_tiles = [&](int k0, int b) {
#pragma unroll
    for (int i = 0; i < 4; ++i) {
      int idx = tid + 256 * i;
      int r = idx >> 3, c4 = (idx & 7) << 2;
      async_ld_b128(X + (size_t)(m0 + r) * CDIM + k0 + c4, &As[b][r][c4]);
    }
#pragma unroll
    for (int i = 0; i < 4; ++i) {
      int idx = tid + 256 * i;
      int r = idx >> 3, c4 = (idx & 7) << 2;
      async_ld_b128(W + (size_t)(n0 + r) * CDIM + k0 + c4, &Ws[b][r][c4]);
    }
  };

  issue_tiles(0, 0);
  int cur = 0;
  for (int k0 = 0; k0 < CDIM; k0 += BK) {
    wait_async0();
    __syncthreads();
    if (k0 + BK < CDIM) issue_tiles(k0 + BK, cur ^ 1);
#pragma unroll
    for (int kk = 0; kk < BK; kk += 4) {
      v2f a0 = *(const v2f*)&As[cur][mrowA][kk + koff];
      v2f a1 = *(const v2f*)&As[cur][mrowA + 16][kk + koff];
#pragma unroll
      for (int nt = 0; nt < 4; ++nt) {
        int n = nbase + nt * 16;
        v2f bfr;
        bfr.x = Ws[cur][n][kk + bka];
        bfr.y = Ws[cur][n][kk + bka + 2];
        acc[0][nt] = wmma4(a0, bfr, acc[0][nt]);
        acc[1][nt] = wmma4(a1, bfr, acc[1][nt]);
      }
    }
    cur ^= 1;
  }
  // C frag layout: VGPR r -> M = r + (lane>=16 ? 8 : 0), N = lane&15
  const int cl = lane & 15;
#pragma unroll
  for (int mi = 0; mi < 2; ++mi) {
    const int rbase = m0 + wm * 32 + mi * 16 + ((lane >> 4) << 3);
#pragma unroll
    for (int nt = 0; nt < 4; ++nt) {
      int col = n0 + wn * 64 + nt * 16 + cl;
#pragma unroll
      for (int r = 0; r < 8; ++r) {
        size_t oidx = (size_t)(rbase + r) * CDIM + col;
        float val = acc[mi][nt][r];
        if (MODE == 1) val = resid[oidx] + val * scale[col];
        Y[oidx] = val;
      }
    }
  }
}

// ---------------------------------------------------------------------------
// KA = K @ A_init  ([16384,1024] x [1024,8])
// ---------------------------------------------------------------------------
__global__ __launch_bounds__(256) void ka_kernel(const float* __restrict__ Kmat,
                                                 const float* __restrict__ A,
                                                 float* __restrict__ KA) {
  __shared__ float red[256];
  const int m = blockIdx.x, tid = threadIdx.x;
  const float* kr = Kmat + (size_t)m * CDIM;
  float acc[RANK] = {};
  for (int c = tid; c < CDIM; c += 256) {
    float kv = kr[c];
#pragma unroll
    for (int r = 0; r < RANK; ++r) acc[r] += kv * A[c * RANK + r];
  }
#pragma unroll
  for (int r = 0; r < RANK; ++r) {
    red[tid] = acc[r]; __syncthreads();
    for (int o = 128; o > 0; o >>= 1) { if (tid < o) red[tid] += red[tid + o]; __syncthreads(); }
    if (tid == 0) KA[(size_t)m * RANK + r] = red[0];
    __syncthreads();
  }
}

// ---------------------------------------------------------------------------
// y0 = K @ W0 (+ KA @ Binit); error = y0 - V; grad = K*error (overwrites V);
// per-block partial sum of error^2 for the loss.
// B tile kept in natural [k][n] layout (async-friendly; no transpose stores).
// ---------------------------------------------------------------------------
__global__ __launch_bounds__(256) void gemm_nn_y0_kernel(const float* __restrict__ Kmat,
                                                         const float* __restrict__ W0,
                                                         const float* __restrict__ KA,
                                                         const float* __restrict__ Binit,
                                                         float* __restrict__ Vgrad,
                                                         float* __restrict__ y0,
                                                         float* __restrict__ losspartial) {
  __shared__ __align__(16) float As[2][BM][LDK];
  __shared__ __align__(16) float Bs[2][BK][LDB];
  __shared__ float red[256];
  const int tid  = threadIdx.x;
  const int lane = tid & 31;
  const int wv   = tid >> 5;
  const int wm   = wv & 3;
  const int wn   = wv >> 2;
  const int bm   = blockIdx.x >> 3;
  const int bn   = blockIdx.x & 7;
  const int m0 = bm * BM, n0 = bn * BN;
  const int mrowA = wm * 32 + (lane & 15);
  const int nbase = wn * 64 + (lane & 15);
  const int koff = (lane < 16) ? 0 : 2;
  const int bka  = (lane < 16) ? 0 : 1;
  v8f acc[2][4] = {};

  auto issue_tiles = [&](int k0, int b) {
#pragma unroll
    for (int i = 0; i < 4; ++i) {
      int idx = tid + 256 * i;
      int r = idx >> 3, c4 = (idx & 7) << 2;
      async_ld_b128(Kmat + (size_t)(m0 + r) * CDIM + k0 + c4, &As[b][r][c4]);
    }
#pragma unroll
    for (int i = 0; i < 4; ++i) {
      int idx = tid + 256 * i;          // 32 k-rows x 32 float4 of n (128 n)
      int kr = idx >> 5, c4 = (idx & 31) << 2;
      async_ld_b128(W0 + (size_t)(k0 + kr) * CDIM + n0 + c4, &Bs[b][kr][c4]);
    }
  };

  issue_tiles(0, 0);
  int cur = 0;
  for (int k0 = 0; k0 < CDIM; k0 += BK) {
    wait_async0();
    __syncthreads();
    if (k0 + BK < CDIM) issue_tiles(k0 + BK, cur ^ 1);
#pragma unroll
    for (int kk = 0; kk < BK; kk += 4) {
      v2f a0 = *(const v2f*)&As[cur][mrowA][kk + koff];
      v2f a1 = *(const v2f*)&As[cur][mrowA + 16][kk + koff];
#pragma unroll
      for (int nt = 0; nt < 4; ++nt) {
        int n = nbase + nt * 16;
        v2f bfr;
        bfr.x = Bs[cur][kk + bka][n];
        bfr.y = Bs[cur][kk + bka + 2][n];
        acc[0][nt] = wmma4(a0, bfr, acc[0][nt]);
        acc[1][nt] = wmma4(a1, bfr, acc[1][nt]);
      }
    }
    cur ^= 1;
  }

  const int cl = lane & 15;
  float sq = 0.f;
#pragma unroll
  for (int mi = 0; mi < 2; ++mi) {
    const int rbase = m0 + wm * 32 + mi * 16 + ((lane >> 4) << 3);
#pragma unroll
    for (int nt = 0; nt < 4; ++nt) {
      int col = n0 + wn * 64 + nt * 16 + cl;
#pragma unroll
      for (int r = 0; r < 8; ++r) {
        int row = rbase + r;
        size_t oidx = (size_t)row * CDIM + col;
        float val = acc[mi][nt][r];
        const float* kar = KA + (size_t)row * RANK;
#pragma unroll
        for (int q = 0; q < RANK; ++q) val += kar[q] * Binit[q * CDIM + col];
        float vv = Vgrad[oidx];          // V
        float err = val - vv;
        float kval = Kmat[oidx];
        y0[oidx] = val;
        Vgrad[oidx] = kval * err;        // gradient overwrites V
        sq += err * err;
      }
    }
  }
  red[tid] = sq; __syncthreads();
  for (int o = 128; o > 0; o >>= 1) { if (tid < o) red[tid] += red[tid + o]; __syncthreads(); }
  if (tid == 0) losspartial[blockIdx.x] = red[0];
}

// ---------------------------------------------------------------------------
// Scan pass 1: per-(b,c) per-chunk sums of gradient (32 chunks of 128 steps)
// ---------------------------------------------------------------------------
__global__ __launch_bounds__(256) void chunksum_kernel(const float* __restrict__ grad,
                                                       float* __restrict__ csum) {
  const int gid = blockIdx.x * 256 + threadIdx.x;   // 0 .. 4*32*1024-1
  const int c  = gid & 1023;
  const int ch = (gid >> 10) & (NCH - 1);
  const int b  = gid >> 15;
  const size_t base = (size_t)b * 4096 * CDIM + (size_t)ch * CLEN * CDIM + c;
  float s = 0.f;
  for (int t = 0; t < CLEN; ++t) s += grad[base + (size_t)t * CDIM];
  csum[gid] = s;
}

// ---------------------------------------------------------------------------
// Scan pass 2: carry = sum of prior chunk sums, then fused elementwise:
// out = (y0 - Q*lr*shifted_cumsum(grad)*pos_scale) * sigmoid(Gl); in-place.
// ---------------------------------------------------------------------------
__global__ __launch_bounds__(256) void scan2_kernel(const float* __restrict__ Q,
                                                    const float* __restrict__ y0,
                                                    const float* __restrict__ Gl,
                                                    const float* __restrict__ log_lr,
                                                    const float* __restrict__ csum,
                                                    float* __restrict__ gradout) {
  const int gid = blockIdx.x * 256 + threadIdx.x;
  const int c  = gid & 1023;
  const int ch = (gid >> 10) & (NCH - 1);   // uniform within a block
  const int b  = gid >> 15;
  float carry = 0.f;
  for (int j = 0; j < ch; ++j) carry += csum[(((b * NCH + j) << 10)) + c];
  const float lrv = fminf(expf(log_lr[c]), 1.0f);
  const size_t base = (size_t)b * 4096 * CDIM + (size_t)ch * CLEN * CDIM + c;
  const int t0 = ch * CLEN;
  for (int i = 0; i < CLEN; ++i) {
    size_t idx = base + (size_t)i * CDIM;
    float g = gradout[idx];
    float shift = carry;
    carry += g;
    float ps = 1.0f / (1.0f + 0.1f * logf((float)(t0 + i + 1)));
    float gate = 1.0f / (1.0f + expf(-Gl[idx]));
    gradout[idx] = (y0[idx] - Q[idx] * lrv * shift * ps) * gate;
  }
}

// ---------------------------------------------------------------------------
// Final loss reduce: GEMM_BLOCKS partials -> d_out[B*T*C] (deterministic tree)
// ---------------------------------------------------------------------------
__global__ __launch_bounds__(256) void loss_kernel(const float* __restrict__ part,
                                                   float* __restrict__ out) {
  __shared__ float red[256];
  const int tid = threadIdx.x;
  float s = 0.f;
  for (int i = tid; i < GEMM_BLOCKS; i += 256) s += part[i];
  red[tid] = s; __syncthreads();
  for (int o = 128; o > 0; o >>= 1) { if (tid < o) red[tid] += red[tid + o]; __syncthreads(); }
  if (tid == 0) out[0] = red[0] * (1.0f / 16777216.0f);
}

// ---------------------------------------------------------------------------
extern "C" void kernel_launch(void* const* d_in, const int* in_sizes, int n_in,
                              void* d_out, int out_size, void* d_ws, size_t ws_size,
                              hipStream_t stream) {
  const float* x      = (const float*)d_in[0];
  const float* W0     = (const float*)d_in[1];
  const float* A_init = (const float*)d_in[2];
  const float* Binit  = (const float*)d_in[3];
  const float* log_lr = (const float*)d_in[4];
  const float* Wq     = (const float*)d_in[5];
  const float* Wk     = (const float*)d_in[6];
  const float* Wv     = (const float*)d_in[7];
  const float* Wo     = (const float*)d_in[8];
  const float* Wg     = (const float*)d_in[9];
  const float* ln_g   = (const float*)d_in[10];
  const float* ln_b   = (const float*)d_in[11];
  const float* ls_g   = (const float*)d_in[12];
  float* out = (float*)d_out;

  const size_t MAT = (size_t)BT_ROWS * CDIM;
  float* ws = (float*)d_ws;
  float* hb   = ws;              // h, later reused as y0
  float* Qb   = ws + 1 * MAT;
  float* Kb   = ws + 2 * MAT;
  float* Vb   = ws + 3 * MAT;    // V -> grad -> gated out
  float* Glb  = ws + 4 * MAT;
  float* KAb  = ws + 5 * MAT;                      // 16384*8
  float* lpb  = KAb + (size_t)BT_ROWS * RANK;      // GEMM_BLOCKS partials
  float* csb  = lpb + GEMM_BLOCKS;                 // 4*32*1024 chunk sums

  dim3 blk(256);
  ln_kernel<<<BT_ROWS, blk, 0, stream>>>(x, ln_g, ln_b, hb);
  gemm_nt_kernel<0><<<GEMM_BLOCKS, blk, 0, stream>>>(hb, Wq, Qb, nullptr, nullptr);
  gemm_nt_kernel<0><<<GEMM_BLOCKS, blk, 0, stream>>>(hb, Wk, Kb, nullptr, nullptr);
  gemm_nt_kernel<0><<<GEMM_BLOCKS, blk, 0, stream>>>(hb, Wv, Vb, nullptr, nullptr);
  gemm_nt_kernel<0><<<GEMM_BLOCKS, blk, 0, stream>>>(x,  Wg, Glb, nullptr, nullptr);
  ka_kernel<<<BT_ROWS, blk, 0, stream>>>(Kb, A_init, KAb);
  gemm_nn_y0_kernel<<<GEMM_BLOCKS, blk, 0, stream>>>(Kb, W0, KAb, Binit, Vb, hb, lpb);
  chunksum_kernel<<<(4 * NCH * 1024) / 256, blk, 0, stream>>>(Vb, csb);
  scan2_kernel<<<(4 * NCH * 1024) / 256, blk, 0, stream>>>(Qb, hb, Glb, log_lr, csb, Vb);
  gemm_nt_kernel<1><<<GEMM_BLOCKS, blk, 0, stream>>>(Vb, Wo, out, x, ls_g);
  loss_kernel<<<1, blk, 0, stream>>>(lpb, out + MAT);
}